// RoIAlign_8169027797167
// MI455X (gfx1250) — compile-verified
//
#include <hip/hip_runtime.h>
#include <cstdint>

typedef float v2f __attribute__((ext_vector_type(2)));
typedef float v8f __attribute__((ext_vector_type(8)));

#define PHB 7
#define PWB 7
#define CCH 256
#define HH 200
#define WW 200
#define SCALEF 0.25f

#define MAXRR 25      // max footprint rows  (roi span <= 24.3 + corner)
#define MAXRC 28      // max footprint cols  (4-aligned origin)
#define FM_PITCH 28   // floats: 112B rows -> 16B aligned

// ---- CDNA5 async global->LDS (ASYNCcnt path) ----
__device__ __forceinline__ void async_b128(uint32_t ldsoff, const float* g) {
  asm volatile("global_load_async_to_lds_b128 %0, %1, off"
               :: "v"(ldsoff), "v"((unsigned long long)(uintptr_t)g) : "memory");
}
__device__ __forceinline__ void wait_async0() {
  asm volatile("s_wait_asynccnt 0" ::: "memory");
}

__device__ __forceinline__ int imin(int a, int b) { return a < b ? a : b; }

__global__ void __launch_bounds__(128)
roialign_wmma_kernel(const float* __restrict__ inp, const float* __restrict__ rois,
                     float* __restrict__ out)
{
  __shared__ __attribute__((aligned(16))) float sFM[16 * MAXRR * FM_PITCH]; // 44800 B
  __shared__ float sV[MAXRC * 8 * 16];                                      // 14336 B
  __shared__ float sAy[16 * 28];                                            // rows 7..15 stay 0
  __shared__ float sAx[16 * 28];
  __shared__ int   sYtab[14][4];   // yl, yh, frac(bits), valid
  __shared__ int   sXtab[14][4];
  __shared__ int   sMM[4];         // ymin, ymax, xmin, xmax
  __shared__ int   sMeta[4];       // y0, RR, x0, RC

  const int roi  = blockIdx.x;
  const int tid  = threadIdx.x;
  const int lane = tid & 31;
  const int wid  = tid >> 5;       // wave id 0..3 (uniform per wave)
  const int m    = lane & 15;      // fragment row (A.M) / col (B.N / D.N)
  const int hif  = lane >> 4;      // half selector for K layout

  if (tid == 0) { sMM[0] = 0x7fffffff; sMM[1] = -0x80000000; sMM[2] = 0x7fffffff; sMM[3] = -0x80000000; }
  for (int i = tid; i < 16*28; i += 128) { sAy[i] = 0.f; sAx[i] = 0.f; }
  __syncthreads();

  const int bidx = (int)rois[roi*5 + 0];

  // ---- Phase A: per-sample corner tables + footprint bbox ----
  if (tid < 28) {
    const float x1 = rois[roi*5+1] * SCALEF;
    const float y1 = rois[roi*5+2] * SCALEF;
    const float x2 = rois[roi*5+3] * SCALEF;
    const float y2 = rois[roi*5+4] * SCALEF;
    const float binw = fmaxf(x2 - x1, 1.f) * (1.f/PWB);
    const float binh = fmaxf(y2 - y1, 1.f) * (1.f/PHB);
    const int isx = tid >= 14;
    const int s   = tid - (isx ? 14 : 0);      // bin*2 + subsample
    const float off = (float)(s >> 1) + ((float)(s & 1) + 0.5f) * 0.5f;
    const float t   = isx ? (x1 + off*binw) : (y1 + off*binh);
    const int   L   = isx ? WW : HH;
    const int valid = (t >= -1.0f && t <= (float)L) ? 1 : 0;
    const float t0  = fmaxf(t, 0.f);
    int lo = (int)floorf(t0); if (lo > L-1) lo = L-1;
    int hc = lo + 1;          if (hc > L-1) hc = L-1;
    const float fr = t0 - (float)lo;
    int* tab = isx ? &sXtab[s][0] : &sYtab[s][0];
    tab[0] = lo; tab[1] = hc; tab[2] = __float_as_int(fr); tab[3] = valid;
    if (valid) {
      if (!isx) { atomicMin(&sMM[0], lo); atomicMax(&sMM[1], hc); }
      else      { atomicMin(&sMM[2], lo); atomicMax(&sMM[3], hc); }
    }
  }
  __syncthreads();
  if (tid == 0) {
    int y0 = (sMM[0] == 0x7fffffff) ? 0 : sMM[0];
    int RR = (sMM[1] == -0x80000000) ? 1 : (sMM[1] - y0 + 1);
    int x0 = (sMM[2] == 0x7fffffff) ? 0 : (sMM[2] & ~3);     // 16B-aligned origin
    int RC = (sMM[3] == -0x80000000) ? 1 : (sMM[3] - x0 + 1);
    sMeta[0] = y0; sMeta[1] = RR; sMeta[2] = x0; sMeta[3] = RC;
  }
  __syncthreads();

  // ---- Build separable weight matrices Ay (7xRR), Ax (7xRC); 0.5*0.5 = sample mean ----
  if (tid < 7) {
    const int y0 = sMeta[0];
    for (int s2 = 0; s2 < 2; ++s2) {
      const int* t = sYtab[tid*2 + s2];
      if (t[3]) {
        const float fr = __int_as_float(t[2]);
        sAy[tid*28 + (t[0] - y0)] += 0.5f * (1.f - fr);
        sAy[tid*28 + (t[1] - y0)] += 0.5f * fr;
      }
    }
  } else if (tid >= 32 && tid < 39) {
    const int pw = tid - 32;
    const int x0 = sMeta[2];
    for (int s2 = 0; s2 < 2; ++s2) {
      const int* t = sXtab[pw*2 + s2];
      if (t[3]) {
        const float fr = __int_as_float(t[2]);
        sAx[pw*28 + (t[0] - x0)] += 0.5f * (1.f - fr);
        sAx[pw*28 + (t[1] - x0)] += 0.5f * fr;
      }
    }
  }
  __syncthreads();

  // Uniform (SGPR) copies so WMMA loops are wave-uniform (EXEC stays all-ones)
  const int y0u = __builtin_amdgcn_readfirstlane(sMeta[0]);
  const int RRu = __builtin_amdgcn_readfirstlane(sMeta[1]);
  const int x0u = __builtin_amdgcn_readfirstlane(sMeta[2]);
  const int RCu = __builtin_amdgcn_readfirstlane(sMeta[3]);
  const int nchunk = (RCu + 3) >> 2;        // 16B chunks per row (<=7)
  const int RRp = (RRu + 3) & ~3;           // K padding stage 1 (<=28)
  const int RCp = (RCu + 3) & ~3;           // K padding stage 2 (<=28)
  const int RRm1 = RRu - 1;
  const int RCm1 = RCu - 1;

  const float* plane0 = inp + (size_t)bidx * CCH * (HH*WW);

  for (int ct = 0; ct < 16; ++ct) {
    // ---- Phase B: async-stage footprint for 16 channels into LDS ----
    const int total = 16 * RRu * nchunk;
    for (int i = tid; i < total; i += 128) {
      const int ch   = i / (RRu * nchunk);
      const int rem  = i - ch * (RRu * nchunk);
      const int row  = rem / nchunk;
      const int cchk = rem - row * nchunk;
      const float* g = plane0 + (size_t)(ct*16 + ch)*(HH*WW)
                              + (size_t)(y0u + row)*WW + x0u + cchk*4;
      const uint32_t lds = (uint32_t)(uintptr_t)&sFM[(ch*MAXRR + row)*FM_PITCH + cchk*4];
      async_b128(lds, g);
    }
    wait_async0();
    __syncthreads();

    // ---- Stage 1: V[x][ph][ch] = sum_y Ay[ph][y] * FM[ch][y][x] ----
    // K-pad handling: Ay cols >= RR are exactly zero, so clamp the B row index
    // (always reads finite staged data) instead of zero-selecting -> branch-free.
    for (int x = wid; x < RCu; x += 4) {
      v8f acc = {};
      for (int kb = 0; kb < RRp; kb += 4) {
        const int k0 = kb + hif*2;                    // A layout: v0=K{0,2}, v1=K{1,3}
        const int r0 = imin(k0,     RRm1);
        const int r1 = imin(k0 + 1, RRm1);
        v2f a, b;
        a.x = sAy[m*28 + k0];
        a.y = sAy[m*28 + k0 + 1];
        b.x = sFM[(m*MAXRR + r0)*FM_PITCH + x];
        b.y = sFM[(m*MAXRR + r1)*FM_PITCH + x];
        acc = __builtin_amdgcn_wmma_f32_16x16x4_f32(false, a, false, b,
                                                    (short)0, acc, false, false);
      }
      if (hif == 0) {                                  // D rows 0..7 live in lanes 0..15
        #pragma unroll
        for (int r = 0; r < 7; ++r)
          sV[(x*8 + r)*16 + m] = acc[r];
      }
    }
    __syncthreads();

    // ---- Stage 2: out[ch][ph][pw] = sum_x Ax[pw][x] * V[x][ph][ch] ----
    for (int ph = wid; ph < 7; ph += 4) {
      v8f acc = {};
      for (int kb = 0; kb < RCp; kb += 4) {
        const int k0 = kb + hif*2;
        const int c0 = imin(k0,     RCm1);
        const int c1 = imin(k0 + 1, RCm1);
        v2f a, b;
        a.x = sAx[m*28 + k0];
        a.y = sAx[m*28 + k0 + 1];
        b.x = sV[(c0*8 + ph)*16 + m];
        b.y = sV[(c1*8 + ph)*16 + m];
        acc = __builtin_amdgcn_wmma_f32_16x16x4_f32(false, a, false, b,
                                                    (short)0, acc, false, false);
      }
      if (hif == 0) {
        float* op = out + (((size_t)roi*CCH + ct*16 + m)*PHB + ph)*PWB;
        #pragma unroll
        for (int r = 0; r < 7; ++r) op[r] = acc[r];    // r == pw
      }
    }
    __syncthreads();
  }
}

extern "C" void kernel_launch(void* const* d_in, const int* in_sizes, int n_in,
                              void* d_out, int out_size, void* d_ws, size_t ws_size,
                              hipStream_t stream) {
  const float* inp  = (const float*)d_in[0];
  const float* rois = (const float*)d_in[1];
  float* out = (float*)d_out;
  const int K = in_sizes[1] / 5;
  (void)n_in; (void)out_size; (void)d_ws; (void)ws_size;
  roialign_wmma_kernel<<<K, 128, 0, stream>>>(inp, rois, out);
}